// ShiftConvGeneral_65738769433039
// MI455X (gfx1250) — compile-verified
//
#include <hip/hip_runtime.h>

// Problem constants from the reference: x is (16, 64, 256, 256) f32, weight (1,1,3,3) f32.
// out[b,c,h,w] = x[b,c,h+dy,w+dx] with zero padding, where (dy,dx) = one-hot argmax
// position of |weight| minus 1. Pure data movement: ~537 MB @ 23.3 TB/s => ~23 us floor.

typedef float v4f __attribute__((ext_vector_type(4)));

// ---------------------------------------------------------------------------
// Kernel 1: tiny device-side argmax of |w| over the 9 taps (jnp.argmax picks
// the FIRST max -> strict '>' comparison). Writes dy, dx into workspace.
// ---------------------------------------------------------------------------
__global__ void shift_argmax_kernel(const float* __restrict__ w,
                                    int* __restrict__ shift) {
  if (blockIdx.x == 0 && threadIdx.x == 0) {
    float best = -1.0f;
    int bi = 0;
#pragma unroll
    for (int i = 0; i < 9; ++i) {
      float a = __builtin_fabsf(w[i]);
      if (a > best) { best = a; bi = i; }
    }
    shift[0] = bi / 3 - 1;  // dy
    shift[1] = bi % 3 - 1;  // dx
  }
}

// ---------------------------------------------------------------------------
// Kernel 2: shifted streaming copy. One float4 per thread.
//   tid -> (row, vec): 64 float4 per 256-wide row; a wave32 covers half a row,
//   so h / dy / dx / all border predicates are wave-uniform (scalar branches).
// Horizontal shift without misaligned b128:
//   load aligned float4 at the output column, plus one neighbor dword
//   (L0-hot: it lives in the adjacent lane's 128B line), recombine in VGPRs.
// NT hints on the bulk load/store: streaming data, keep the 192MB L2 clean.
// ---------------------------------------------------------------------------
__global__ __launch_bounds__(256) void shift_copy_kernel(
    const float* __restrict__ x, float* __restrict__ out,
    const int* __restrict__ shift) {
  constexpr int H = 256, W = 256;

  // Uniform loads from kernarg-derived restrict pointer -> SMEM (s_load).
  const int dy = shift[0];
  const int dx = shift[1];

  const size_t tid   = (size_t)blockIdx.x * 256u + threadIdx.x;
  const size_t row   = tid >> 6;                 // global row index: (b*C + c)*H + h
  const int    vec   = (int)(tid & 63);          // float4 slot within the row
  const int    h     = (int)(row & (size_t)(H - 1));
  const size_t plane = row >> 8;                 // b*C + c   (H == 256)
  const int    w0    = vec << 2;                 // output column (16B aligned)

  v4f r = {0.0f, 0.0f, 0.0f, 0.0f};

  const int sh = h + dy;                          // wave-uniform
  if (sh >= 0 && sh < H) {
    const float* __restrict__ src = x + (plane * (size_t)H + (size_t)sh) * (size_t)W;

    // Aligned 16B streaming load at the output column.
    v4f v = __builtin_nontemporal_load((const v4f*)(src + w0));

    if (dx == 0) {
      r = v;
    } else if (dx > 0) {                          // need src cols [w0+1 .. w0+4]
      float e = 0.0f;
      if (w0 + 4 < W) e = src[w0 + 4];            // wave-mate's line: L0 hit
      r = (v4f){v.y, v.z, v.w, e};
    } else {                                      // dx == -1: src cols [w0-1 .. w0+2]
      float e = 0.0f;
      if (w0 >= 4) e = src[w0 - 1];               // wave-mate's line: L0 hit
      r = (v4f){e, v.x, v.y, v.z};
    }
  }

  __builtin_nontemporal_store(r, (v4f*)(out + row * (size_t)W + w0));
}

// ---------------------------------------------------------------------------
// Launch: argmax (1 thread) then the shifted copy (stream-ordered, so the
// ws-resident shift is visible to kernel 2; graph-capture safe).
// ---------------------------------------------------------------------------
extern "C" void kernel_launch(void* const* d_in, const int* in_sizes, int n_in,
                              void* d_out, int out_size, void* d_ws, size_t ws_size,
                              hipStream_t stream) {
  const float* x = (const float*)d_in[0];   // (16,64,256,256) f32
  const float* w = (const float*)d_in[1];   // (1,1,3,3) f32
  float* out = (float*)d_out;
  int* shift = (int*)d_ws;                  // 2 ints of scratch

  shift_argmax_kernel<<<1, 32, 0, stream>>>(w, shift);

  const size_t total_vec = (size_t)out_size / 4;          // 16,777,216 float4s
  const int block = 256;
  const int grid  = (int)((total_vec + (size_t)block - 1) / (size_t)block);
  shift_copy_kernel<<<grid, block, 0, stream>>>(x, out, shift);
}